// ImprovedActorCritic_53051436040194
// MI455X (gfx1250) — compile-verified
//
#include <hip/hip_runtime.h>
#include <hip/hip_bf16.h>

typedef __attribute__((ext_vector_type(2))) float v2f;
typedef __attribute__((ext_vector_type(8))) float v8f;

#define NB_RED 512
#define RED_THREADS 256

// ---------------- Kernel 1: masked reduction of coordinates ----------------
__global__ void ac_reduce_kernel(const float* __restrict__ coords,
                                 const unsigned char* __restrict__ visited,
                                 float* __restrict__ partials, int N)
{
    __shared__ float sx[RED_THREADS], sy[RED_THREADS], sc[RED_THREADS];
    const float2* __restrict__ c2 = (const float2*)coords;
    const int tid = threadIdx.x;
    float ax = 0.f, ay = 0.f, ac = 0.f;
    for (int i = blockIdx.x * RED_THREADS + tid; i < N; i += NB_RED * RED_THREADS) {
        const float2 c = c2[i];                       // one global_load_b64
        const float a = visited[i] ? 0.f : 1.f;       // branchless mask
        ax = fmaf(a, c.x, ax);
        ay = fmaf(a, c.y, ay);
        ac += a;
    }
    sx[tid] = ax; sy[tid] = ay; sc[tid] = ac;
    __syncthreads();
    for (int s = RED_THREADS / 2; s > 0; s >>= 1) {
        if (tid < s) {
            sx[tid] += sx[tid + s];
            sy[tid] += sy[tid + s];
            sc[tid] += sc[tid + s];
        }
        __syncthreads();
    }
    if (tid == 0) {
        partials[3 * blockIdx.x + 0] = sx[0];
        partials[3 * blockIdx.x + 1] = sy[0];
        partials[3 * blockIdx.x + 2] = sc[0];
    }
}

// ---------------- WMMA batch-1 matvec: y[0:512] = relu?(x[0:K] @ W + b) ----
// W is K x 512 row-major. 32 waves (1024 threads), one 16-col tile per wave.
// A-matrix (16x4 f32): row M=0 carries x, rows 1..15 are zero. Per the CDNA5
// 16x4 f32 A layout lane 0 must hold K=k..k+1 and lane 16 K=k+2..k+3; we build
// this BRANCHLESSLY: every lane does a uniform ds_load_b64 of x[k+xoff..+1]
// and scales by sel (1 only on lanes 0 and 16). B (4x16): b.x/.y = rows
// K=kk,kk+1 at column col. Result row M=0 lands in acc[0] of lanes 0-15.
__device__ __forceinline__ void matvec512_wmma(const float* __restrict__ W,
                                               const float* __restrict__ bias,
                                               const float* __restrict__ x,
                                               float* __restrict__ y,
                                               int K, bool do_relu)
{
    const int tid  = threadIdx.x;
    const int lane = tid & 31;
    const int col  = ((tid >> 5) << 4) + (lane & 15);      // wave*16 + n
    const int xoff = (lane >> 4) << 1;                     // 0 (lo half) / 2 (hi half)
    const float sel = ((lane & 15) == 0) ? 1.f : 0.f;      // lanes 0 and 16 only
    v8f acc = {};
    for (int k = 0; k < K; k += 4) {
        const v2f xv = *(const v2f*)(x + k + xoff);        // uniform ds_load_b64
        v2f a;
        a.x = xv.x * sel;
        a.y = xv.y * sel;
        const int kk = k + xoff;
        v2f b;
        b.x = W[kk * 512 + col];
        b.y = W[kk * 512 + 512 + col];
        acc = __builtin_amdgcn_wmma_f32_16x16x4_f32(
            /*neg_a=*/false, a, /*neg_b=*/false, b,
            /*c_mod=*/(short)0, acc, /*reuse_a=*/false, /*reuse_b=*/false);
    }
    if (lane < 16) {
        float r = acc[0] + bias[col];
        if (do_relu) r = fmaxf(r, 0.f);
        y[col] = r;
    }
    __syncthreads();
}

__device__ __forceinline__ float block_sum_512(float v, float* red)
{
    const int tid = threadIdx.x;
    if (tid < 512) red[tid] = v;
    __syncthreads();
    for (int s = 256; s > 0; s >>= 1) {
        if (tid < s) red[tid] += red[tid + s];
        __syncthreads();
    }
    float r = red[0];
    __syncthreads();
    return r;
}

// ---------------- Kernel 2: the whole batch-1 MLP in one workgroup ---------
__global__ void __launch_bounds__(1024)
ac_mlp_kernel(const float* __restrict__ coords, const int* __restrict__ cur_idx_p,
              const float* __restrict__ Wemb, const float* __restrict__ bemb,
              const float* __restrict__ Wc1,  const float* __restrict__ bc1,
              const float* __restrict__ Wc2,  const float* __restrict__ bc2,
              const float* __restrict__ Wv1,  const float* __restrict__ bv1,
              const float* __restrict__ Wv2,  const float* __restrict__ bv2,
              const float* __restrict__ partials,
              float* __restrict__ consts, float* __restrict__ out, int N)
{
    __shared__ float ctx[1024];
    __shared__ float h1[512];
    __shared__ float ce[512];
    __shared__ float v1s[512];
    __shared__ float red[512];

    const int tid = threadIdx.x;

    // Reduce the 512 per-block partial triples.
    float psx = 0.f, psy = 0.f, pc = 0.f;
    if (tid < NB_RED) {
        psx = partials[3 * tid + 0];
        psy = partials[3 * tid + 1];
        pc  = partials[3 * tid + 2];
    }
    const float sumx = block_sum_512(psx, red);
    const float sumy = block_sum_512(psy, red);
    const float cnt  = block_sum_512(pc,  red);

    const int   idx = cur_idx_p[0];
    const float c0  = coords[2 * idx], c1 = coords[2 * idx + 1];
    const bool  any = (cnt > 0.f);
    const float inv = any ? (1.f / cnt) : 0.f;
    const float mx  = sumx * inv, my = sumy * inv;

    // context = [current_embed ; masked-mean embed], via the rank-2 factorization
    if (tid < 512) {
        const float w0 = Wemb[tid], w1 = Wemb[512 + tid], bb = bemb[tid];
        ctx[tid] = fmaf(c0, w0, fmaf(c1, w1, bb));
        const float m = fmaf(mx, w0, fmaf(my, w1, bb));
        ctx[512 + tid] = any ? m : 0.f;
    }
    __syncthreads();

    matvec512_wmma(Wc1, bc1, ctx, h1, 1024, true);  // h  = relu(ctx @ Wc1 + bc1)
    matvec512_wmma(Wc2, bc2, h1,  ce,  512, true);  // ce = relu(h @ Wc2 + bc2)
    matvec512_wmma(Wv1, bv1, ce,  v1s, 512, true);  // v1 = relu(ce @ Wv1 + bv1)

    // u = W_emb @ ce (2 dots), cb = b_emb . ce, state_value = v1 . Wv2 + bv2
    float pu0 = 0.f, pu1 = 0.f, pcb = 0.f, psv = 0.f;
    if (tid < 512) {
        const float e = ce[tid];
        pu0 = Wemb[tid] * e;
        pu1 = Wemb[512 + tid] * e;
        pcb = bemb[tid] * e;
        psv = v1s[tid] * Wv2[tid];
    }
    const float u0 = block_sum_512(pu0, red);
    const float u1 = block_sum_512(pu1, red);
    const float cb = block_sum_512(pcb, red);
    const float sv = block_sum_512(psv, red);

    if (tid == 0) {
        consts[0] = u0;
        consts[1] = u1;
        consts[2] = cb;
        out[N] = sv + bv2[0];   // state_value appended after the N logits
    }
}

// ---------------- Kernel 3: streaming masked scoring -----------------------
__global__ void ac_score_kernel(const float* __restrict__ coords,
                                const unsigned char* __restrict__ visited,
                                const float* __restrict__ consts,
                                float* __restrict__ out, int N)
{
    const int i = blockIdx.x * blockDim.x + threadIdx.x;
    if (i >= N) return;
    const float2 c = ((const float2*)coords)[i];          // one global_load_b64
    const float u0 = consts[0], u1 = consts[1], cb = consts[2];
    const float v  = fmaf(c.x, u0, fmaf(c.y, u1, cb));
    out[i] = visited[i] ? -__builtin_inff() : v;
}

extern "C" void kernel_launch(void* const* d_in, const int* in_sizes, int n_in,
                              void* d_out, int out_size, void* d_ws, size_t ws_size,
                              hipStream_t stream)
{
    (void)n_in; (void)out_size; (void)ws_size;
    const float*         coords  = (const float*)d_in[0];
    const int*           cur_idx = (const int*)d_in[1];
    const unsigned char* visited = (const unsigned char*)d_in[2];  // jax bool = 1 byte
    const float* Wemb = (const float*)d_in[3];
    const float* bemb = (const float*)d_in[4];
    const float* Wc1  = (const float*)d_in[5];
    const float* bc1  = (const float*)d_in[6];
    const float* Wc2  = (const float*)d_in[7];
    const float* bc2  = (const float*)d_in[8];
    const float* Wv1  = (const float*)d_in[9];
    const float* bv1  = (const float*)d_in[10];
    const float* Wv2  = (const float*)d_in[11];
    const float* bv2  = (const float*)d_in[12];
    float* out = (float*)d_out;

    const int N = in_sizes[0] / 2;

    float* ws       = (float*)d_ws;
    float* partials = ws;                 // 3 * NB_RED floats
    float* consts   = ws + 3 * NB_RED;    // u0, u1, cb

    ac_reduce_kernel<<<NB_RED, RED_THREADS, 0, stream>>>(coords, visited, partials, N);
    ac_mlp_kernel<<<1, 1024, 0, stream>>>(coords, cur_idx, Wemb, bemb,
                                          Wc1, bc1, Wc2, bc2, Wv1, bv1, Wv2, bv2,
                                          partials, consts, out, N);
    ac_score_kernel<<<(N + 255) / 256, 256, 0, stream>>>(coords, visited, consts, out, N);
}